// EdgeConvBlockV2_21741124452963
// MI455X (gfx1250) — compile-verified
//
#include <hip/hip_runtime.h>

#define B_ 8
#define C_ 64
#define N_ 4096
#define K_ 20
#define BN_EPS 1e-5f

typedef float v2f __attribute__((ext_vector_type(2)));
typedef float v8f __attribute__((ext_vector_type(8)));

// ---------------------------------------------------------------------------
// Kernel 1: 1x1 convs (local+edge combined, edge) stored point-major + sq norms
// grid = B * (N/256), block = 256
// ---------------------------------------------------------------------------
__global__ __launch_bounds__(256) void k1_conv(
    const float* __restrict__ feat, const float* __restrict__ W1,
    const float* __restrict__ W2, float* __restrict__ LEt,
    float* __restrict__ Et, float* __restrict__ sqg) {
  __shared__ float Wle[64][64];
  __shared__ float We[64][64];
  int b  = blockIdx.x >> 4;          // 16 blocks per batch
  int n0 = (blockIdx.x & 15) << 8;   // 256 points per block
  for (int i = threadIdx.x; i < 4096; i += 256) {
    float w2 = W2[i];
    Wle[i >> 6][i & 63] = W1[i] + w2;
    We[i >> 6][i & 63]  = w2;
  }
  __syncthreads();
  int n = n0 + threadIdx.x;
  const float* fp = feat + (size_t)b * C_ * N_ + n;
  float f[64];
  float sq = 0.f;
#pragma unroll
  for (int c = 0; c < 64; ++c) { f[c] = fp[c * N_]; sq = fmaf(f[c], f[c], sq); }
  sqg[b * N_ + n] = sq;
  float* lep = LEt + ((size_t)b * N_ + n) * 64;
  float* ep  = Et  + ((size_t)b * N_ + n) * 64;
  for (int o = 0; o < 64; ++o) {
    float le = 0.f, e = 0.f;
#pragma unroll
    for (int c = 0; c < 64; ++c) {
      le = fmaf(Wle[o][c], f[c], le);
      e  = fmaf(We[o][c],  f[c], e);
    }
    lep[o] = le;
    ep[o]  = e;
  }
}

// ---------------------------------------------------------------------------
// Kernel 2: fused Gram-matrix (f32 WMMA) + streaming top-20 nearest neighbors
// grid = B * (N/128), block = 128 (4 waves; wave w owns rows [w*32, w*32+32))
// key = sq[m] - 2*dot(f_n, f_m)  (row-constant sq[n] dropped: order-invariant)
// ---------------------------------------------------------------------------
__global__ __launch_bounds__(128) void k2_knn(
    const float* __restrict__ feat, const float* __restrict__ sqg,
    int* __restrict__ knn) {
  __shared__ float Dt[128][65];      // pitch 65 -> conflict-free row scans
  int b    = blockIdx.x >> 5;        // 32 blocks per batch
  int n0   = (blockIdx.x & 31) << 7; // 128 rows per block
  int lane = threadIdx.x & 31;
  int wave = threadIdx.x >> 5;
  int lm   = lane & 15;
  int khalf = (lane >> 4) << 1;      // lanes 16-31 hold K+2, K+3

  const float* fb  = feat + (size_t)b * C_ * N_;
  const float* sqb = sqg + b * N_;

  // A fragments (16x4 f32 tiles of f^T for this wave's 32 rows), kept resident
  v2f afrag[2][16];
#pragma unroll
  for (int rb = 0; rb < 2; ++rb) {
    int row = n0 + wave * 32 + rb * 16 + lm;
#pragma unroll
    for (int kk = 0; kk < 16; ++kk) {
      int ch = kk * 4 + khalf;
      afrag[rb][kk][0] = fb[ch * N_ + row];
      afrag[rb][kk][1] = fb[(ch + 1) * N_ + row];
    }
  }

  float bestv[K_];
  int   besti[K_];
#pragma unroll
  for (int i = 0; i < K_; ++i) { bestv[i] = 3.4e38f; besti[i] = 0; }

  for (int m0 = 0; m0 < N_; m0 += 64) {
    v8f acc[2][4];
#pragma unroll
    for (int rb = 0; rb < 2; ++rb)
#pragma unroll
      for (int cb = 0; cb < 4; ++cb)
#pragma unroll
        for (int v = 0; v < 8; ++v) acc[rb][cb][v] = 0.f;

#pragma unroll
    for (int kk = 0; kk < 16; ++kk) {
      int ch = kk * 4 + khalf;
      v2f bf[4];
#pragma unroll
      for (int cb = 0; cb < 4; ++cb) {
        int col = m0 + cb * 16 + lm;
        bf[cb][0] = fb[ch * N_ + col];
        bf[cb][1] = fb[(ch + 1) * N_ + col];
      }
#pragma unroll
      for (int rb = 0; rb < 2; ++rb)
#pragma unroll
        for (int cb = 0; cb < 4; ++cb)
          acc[rb][cb] = __builtin_amdgcn_wmma_f32_16x16x4_f32(
              false, afrag[rb][kk], false, bf[cb], (short)0, acc[rb][cb],
              false, false);
    }

    // spill C fragments to LDS (C layout: vgpr v -> row v / v+8, col = lane&15)
#pragma unroll
    for (int rb = 0; rb < 2; ++rb)
#pragma unroll
      for (int cb = 0; cb < 4; ++cb)
#pragma unroll
        for (int v = 0; v < 8; ++v) {
          int M = wave * 32 + rb * 16 + v + ((lane >= 16) ? 8 : 0);
          Dt[M][cb * 16 + lm] = acc[rb][cb][v];
        }
    __syncthreads();

    // thread r owns row r: register-resident sorted top-20 insertion
    {
      int r = threadIdx.x;
      for (int c = 0; c < 64; ++c) {
        int m = m0 + c;
        float cand = sqb[m] - 2.0f * Dt[r][c];
        if (cand < bestv[K_ - 1]) {
          float v = cand;
          int   id = m;
#pragma unroll
          for (int i = 0; i < K_; ++i) {
            if (v < bestv[i]) {
              float tv = bestv[i]; int ti = besti[i];
              bestv[i] = v; besti[i] = id;
              v = tv; id = ti;
            }
          }
        }
      }
    }
    __syncthreads();
  }

  int n = n0 + threadIdx.x;
  int* kp = knn + ((size_t)b * N_ + n) * K_;
#pragma unroll
  for (int i = 0; i < K_; ++i) kp[i] = besti[i];
}

// ---------------------------------------------------------------------------
// Kernel 3: gather neighbors, per-channel sum/sumsq (f64 atomics) + max/min
// grid = B * (N/64), block = 256 (64 points x 4 channel-groups of 16)
// ---------------------------------------------------------------------------
__global__ __launch_bounds__(256) void k3_stats(
    const float* __restrict__ LEt, const float* __restrict__ Et,
    const int* __restrict__ knn, float* __restrict__ Mx,
    float* __restrict__ Mn, double* __restrict__ sums) {
  __shared__ float ssum[64];
  __shared__ float ssum2[64];
  if (threadIdx.x < 64) { ssum[threadIdx.x] = 0.f; ssum2[threadIdx.x] = 0.f; }
  __syncthreads();
  int b  = blockIdx.x >> 6;
  int n0 = (blockIdx.x & 63) << 6;
  int p  = threadIdx.x >> 2;
  int q  = threadIdx.x & 3;
  int n  = n0 + p;
  int c0 = q * 16;
  const float* lep = LEt + (((size_t)b * N_ + n) * 64 + c0);
  float le[16];
#pragma unroll
  for (int i = 0; i < 16; ++i) le[i] = lep[i];
  float s[16], s2[16], mx[16], mn[16];
#pragma unroll
  for (int i = 0; i < 16; ++i) { s[i] = 0.f; s2[i] = 0.f; mx[i] = -3.4e38f; mn[i] = 3.4e38f; }
  const int* kp = knn + ((size_t)b * N_ + n) * K_;
  for (int j = 0; j < K_; ++j) {
    int m = kp[j];
    const float* ep = Et + (((size_t)b * N_ + m) * 64 + c0);
#pragma unroll
    for (int i = 0; i < 16; ++i) {
      float x = le[i] - ep[i];
      s[i] += x;
      s2[i] = fmaf(x, x, s2[i]);
      mx[i] = fmaxf(mx[i], x);
      mn[i] = fminf(mn[i], x);
    }
  }
  float* mxp = Mx + (((size_t)b * N_ + n) * 64 + c0);
  float* mnp = Mn + (((size_t)b * N_ + n) * 64 + c0);
#pragma unroll
  for (int i = 0; i < 16; ++i) {
    mxp[i] = mx[i];
    mnp[i] = mn[i];
    atomicAdd(&ssum[c0 + i], s[i]);
    atomicAdd(&ssum2[c0 + i], s2[i]);
  }
  __syncthreads();
  if (threadIdx.x < 64) {
    atomicAdd(&sums[threadIdx.x], (double)ssum[threadIdx.x]);
    atomicAdd(&sums[64 + threadIdx.x], (double)ssum2[threadIdx.x]);
  }
}

// ---------------------------------------------------------------------------
// Kernel 4a: fold BN into per-channel scale/bias.  1 block, 64 threads.
// ---------------------------------------------------------------------------
__global__ void k4a_scale(const double* __restrict__ sums,
                          const float* __restrict__ gamma,
                          const float* __restrict__ beta,
                          float* __restrict__ sb) {
  int c = threadIdx.x;
  double cnt  = (double)B_ * N_ * K_;
  double mean = sums[c] / cnt;
  double var  = sums[64 + c] / cnt - mean * mean;
  float  scale = gamma[c] * rsqrtf((float)var + BN_EPS);
  sb[c]      = scale;
  sb[64 + c] = beta[c] - (float)mean * scale;
}

// ---------------------------------------------------------------------------
// Kernel 4b: out = relu(scale * (scale>=0 ? max : min) + bias), (B,C,N) layout
// ---------------------------------------------------------------------------
__global__ __launch_bounds__(256) void k4b_final(
    const float* __restrict__ Mx, const float* __restrict__ Mn,
    const float* __restrict__ sb, float* __restrict__ out) {
  int idx = blockIdx.x * 256 + threadIdx.x;  // (b*64 + c)*4096 + n
  int n = idx & (N_ - 1);
  int c = (idx >> 12) & 63;
  int b = idx >> 18;
  float scale = sb[c], bias = sb[64 + c];
  size_t t = ((size_t)b * N_ + n) * 64 + c;
  float v = (scale >= 0.f) ? Mx[t] : Mn[t];
  out[idx] = fmaxf(fmaf(v, scale, bias), 0.f);
}

// ---------------------------------------------------------------------------
extern "C" void kernel_launch(void* const* d_in, const int* in_sizes, int n_in,
                              void* d_out, int out_size, void* d_ws,
                              size_t ws_size, hipStream_t stream) {
  const float* feat  = (const float*)d_in[0];
  const float* W1    = (const float*)d_in[1];
  const float* W2    = (const float*)d_in[2];
  const float* gamma = (const float*)d_in[3];
  const float* beta  = (const float*)d_in[4];
  (void)in_sizes; (void)n_in; (void)out_size; (void)ws_size;

  char* w = (char*)d_ws;
  const size_t PTS = (size_t)B_ * N_;            // 32768 points
  float*  LEt  = (float*)w;                       w += PTS * 64 * sizeof(float); // 8MB
  float*  Et   = (float*)w;                       w += PTS * 64 * sizeof(float); // 8MB
  float*  Mx   = (float*)w;                       w += PTS * 64 * sizeof(float); // 8MB
  float*  Mn   = (float*)w;                       w += PTS * 64 * sizeof(float); // 8MB
  float*  sqg  = (float*)w;                       w += PTS * sizeof(float);      // 128KB
  int*    knn  = (int*)w;                         w += PTS * K_ * sizeof(int);   // 2.5MB
  double* sums = (double*)w;                      w += 128 * sizeof(double);
  float*  sb   = (float*)w;                       w += 128 * sizeof(float);

  k1_conv<<<B_ * (N_ / 256), 256, 0, stream>>>(feat, W1, W2, LEt, Et, sqg);
  k2_knn<<<B_ * (N_ / 128), 128, 0, stream>>>(feat, sqg, knn);
  hipMemsetAsync(sums, 0, 128 * sizeof(double), stream);
  k3_stats<<<B_ * (N_ / 64), 256, 0, stream>>>(LEt, Et, knn, Mx, Mn, sums);
  k4a_scale<<<1, 64, 0, stream>>>(sums, gamma, beta, sb);
  k4b_final<<<(B_ * 64 * N_) / 256, 256, 0, stream>>>(Mx, Mn, sb, (float*)d_out);
}